// SimpleGraphResidual_31980326486703
// MI455X (gfx1250) — compile-verified
//
#include <hip/hip_runtime.h>

#define HIDDIM   64
#define HID4     16          // float4 chunks per node row
#define TILE     256         // edges staged per block iteration
#define ALPHA_C  0.1f
#define COEF_C   0.3f        // (1-ALPHA)/K = 0.9/3

#if defined(__has_builtin)
#  if __has_builtin(__builtin_amdgcn_global_load_async_to_lds_b32) && \
      __has_builtin(__builtin_amdgcn_s_wait_asynccnt)
#    define HAVE_ASYNC_LDS 1
#  else
#    define HAVE_ASYNC_LDS 0
#  endif
#else
#  define HAVE_ASYNC_LDS 0
#endif

typedef __attribute__((address_space(1))) int* gptr_i32;
typedef __attribute__((address_space(3))) int* lptr_i32;

// ---------------- degree / norm precompute ----------------

__global__ void sgr_zero_cnt(int* __restrict__ cnt, int N) {
    int i = blockIdx.x * blockDim.x + threadIdx.x;
    if (i < N) cnt[i] = 0;
}

__global__ void sgr_count(const int* __restrict__ dst, int* __restrict__ cnt, int E) {
    int e = blockIdx.x * blockDim.x + threadIdx.x;
    if (e < E) atomicAdd(&cnt[dst[e]], 1);
}

__global__ void sgr_dinv(const int* __restrict__ cnt,
                         float* __restrict__ dinv, float* __restrict__ invc, int N) {
    int i = blockIdx.x * blockDim.x + threadIdx.x;
    if (i < N) {
        int c = cnt[i];
        dinv[i] = rsqrtf((float)(c + 1));           // deg includes self loop, always > 0
        invc[i] = 1.0f / (float)(c > 1 ? c : 1);    // mean divisor: max(cnt, 1)
    }
}

// ---------------- per-step elementwise kernels ----------------

// acc = dinv ⊙ h   (self-loop inner contribution; edge sums accumulate on top)
__global__ void sgr_init_acc(const float4* __restrict__ h, const float* __restrict__ dinv,
                             float4* __restrict__ acc, int total4) {
    int i = blockIdx.x * blockDim.x + threadIdx.x;
    if (i < total4) {
        float s = dinv[i >> 4];                     // 16 float4 per node
        float4 v = h[i];
        v.x *= s; v.y *= s; v.z *= s; v.w *= s;
        acc[i] = v;
    }
}

// h_new = dinv ⊙ acc ; out = (first ? alpha*x : out) + coef*h_new
__global__ void sgr_finish(const float4* __restrict__ acc, const float* __restrict__ dinv,
                           const float4* __restrict__ x, float4* __restrict__ h,
                           float4* __restrict__ out, int total4, int first) {
    int i = blockIdx.x * blockDim.x + threadIdx.x;
    if (i < total4) {
        float s = dinv[i >> 4];
        float4 a = acc[i];
        a.x *= s; a.y *= s; a.z *= s; a.w *= s;
        h[i] = a;
        float4 o;
        if (first) {
            float4 xv = x[i];
            o.x = ALPHA_C * xv.x; o.y = ALPHA_C * xv.y;
            o.z = ALPHA_C * xv.z; o.w = ALPHA_C * xv.w;
        } else {
            o = out[i];
        }
        o.x += COEF_C * a.x; o.y += COEF_C * a.y;
        o.z += COEF_C * a.z; o.w += COEF_C * a.w;
        out[i] = o;
    }
}

// h = relu(out) ; acc = 0 (re-used as mean accumulator)
__global__ void sgr_relu_zero(const float4* __restrict__ out, float4* __restrict__ h,
                              float4* __restrict__ acc, int total4) {
    int i = blockIdx.x * blockDim.x + threadIdx.x;
    if (i < total4) {
        float4 o = out[i];
        o.x = fmaxf(o.x, 0.0f); o.y = fmaxf(o.y, 0.0f);
        o.z = fmaxf(o.z, 0.0f); o.w = fmaxf(o.w, 0.0f);
        h[i] = o;
        acc[i] = make_float4(0.0f, 0.0f, 0.0f, 0.0f);
    }
}

// out = relu(acc * invc + x)
__global__ void sgr_final(const float4* __restrict__ acc, const float* __restrict__ invc,
                          const float4* __restrict__ x, float4* __restrict__ out, int total4) {
    int i = blockIdx.x * blockDim.x + threadIdx.x;
    if (i < total4) {
        float s = invc[i >> 4];
        float4 a = acc[i];
        float4 xv = x[i];
        a.x = fmaxf(a.x * s + xv.x, 0.0f);
        a.y = fmaxf(a.y * s + xv.y, 0.0f);
        a.z = fmaxf(a.z * s + xv.z, 0.0f);
        a.w = fmaxf(a.w * s + xv.w, 0.0f);
        out[i] = a;
    }
}

// ---------------- edge scatter: acc[dst] += (sc ? sc[src] : 1) * h[src] -------
// 256 threads = 16 groups of 16 lanes; each group handles one edge at a time,
// 16 lanes x float4 = one 64-float row (256B fully coalesced). Edge indices are
// staged into LDS via gfx1250 async global->LDS copies (ASYNCcnt path).
// sc == dinv for the propagation steps (per-edge norm factored as
// dinv[dst]*(dinv[src]*h[src]), outer factor applied in sgr_finish);
// sc == nullptr for the plain mean aggregation.

__global__ void sgr_scatter(const int* __restrict__ src, const int* __restrict__ dst,
                            const float4* __restrict__ h, const float* __restrict__ sc,
                            float* __restrict__ acc, int E) {
    __shared__ int s_src[TILE];
    __shared__ int s_dst[TILE];
    const int tid    = threadIdx.x;
    const int group  = tid >> 4;   // 0..15
    const int lane16 = tid & 15;   // 0..15

    for (int base = blockIdx.x * TILE; base < E; base += gridDim.x * TILE) {
        int n = E - base; if (n > TILE) n = TILE;

        if (tid < n) {
#if HAVE_ASYNC_LDS
            __builtin_amdgcn_global_load_async_to_lds_b32(
                (gptr_i32)(src + base + tid),
                (lptr_i32)&s_src[tid], 0, 0);
            __builtin_amdgcn_global_load_async_to_lds_b32(
                (gptr_i32)(dst + base + tid),
                (lptr_i32)&s_dst[tid], 0, 0);
#else
            s_src[tid] = src[base + tid];
            s_dst[tid] = dst[base + tid];
#endif
        }
#if HAVE_ASYNC_LDS
        __builtin_amdgcn_s_wait_asynccnt(0);   // each wave waits for its own copies
#endif
        __syncthreads();

        for (int e = group; e < n; e += 16) {
            int s = s_src[e];                       // LDS broadcast within group
            int d = s_dst[e];
            float4 v = h[(size_t)s * HID4 + lane16];
            if (sc) {                               // uniform branch
                float f = sc[s];                    // group-uniform dword (L2 broadcast)
                v.x *= f; v.y *= f; v.z *= f; v.w *= f;
            }
            float* a = acc + (size_t)d * HIDDIM + lane16 * 4;
            unsafeAtomicAdd(a + 0, v.x);            // -> global_atomic_add_f32 (no return)
            unsafeAtomicAdd(a + 1, v.y);
            unsafeAtomicAdd(a + 2, v.z);
            unsafeAtomicAdd(a + 3, v.w);
        }
        __syncthreads();
    }
}

// ---------------- host-side launcher ----------------

extern "C" void kernel_launch(void* const* d_in, const int* in_sizes, int n_in,
                              void* d_out, int out_size, void* d_ws, size_t ws_size,
                              hipStream_t stream) {
    const float* x  = (const float*)d_in[0];
    const int*   ei = (const int*)d_in[1];      // [2, E] flat: src row then dst row
    const int NH = in_sizes[0];                 // N * 64
    const int N  = NH / HIDDIM;
    const int E  = in_sizes[1] / 2;
    const int* src = ei;
    const int* dst = ei + E;

    // workspace layout (all offsets 16B aligned: N*4 = 400000 = 16*25000)
    char*  ws   = (char*)d_ws;
    int*   cnt  = (int*)ws;
    float* dinv = (float*)(ws + (size_t)N * 4);
    float* invc = (float*)(ws + (size_t)2 * N * 4);
    float* acc  = (float*)(ws + (size_t)3 * N * 4);
    float* h    = acc + (size_t)N * HIDDIM;
    float* out  = (float*)d_out;

    const int total4 = NH / 4;
    const dim3 blk(256);
    const int gN = (N + 255) / 256;
    const int gE = (E + 255) / 256;
    const int g4 = (total4 + 255) / 256;
    const int gT = (E + TILE - 1) / TILE;

    sgr_zero_cnt<<<gN, blk, 0, stream>>>(cnt, N);
    sgr_count   <<<gE, blk, 0, stream>>>(dst, cnt, E);
    sgr_dinv    <<<gN, blk, 0, stream>>>(cnt, dinv, invc, N);

    const float* hin = x;
    for (int k = 0; k < 3; ++k) {
        sgr_init_acc<<<g4, blk, 0, stream>>>((const float4*)hin, dinv,
                                             (float4*)acc, total4);
        sgr_scatter <<<gT, blk, 0, stream>>>(src, dst, (const float4*)hin, dinv, acc, E);
        sgr_finish  <<<g4, blk, 0, stream>>>((const float4*)acc, dinv, (const float4*)x,
                                             (float4*)h, (float4*)out, total4, k == 0);
        hin = h;
    }

    sgr_relu_zero<<<g4, blk, 0, stream>>>((const float4*)out, (float4*)h,
                                          (float4*)acc, total4);
    sgr_scatter  <<<gT, blk, 0, stream>>>(src, dst, (const float4*)h, nullptr, acc, E);
    sgr_final    <<<g4, blk, 0, stream>>>((const float4*)acc, invc, (const float4*)x,
                                          (float4*)out, total4);
}